// RobustListLearner_32384053411867
// MI455X (gfx1250) — compile-verified
//
#include <hip/hip_runtime.h>

typedef __attribute__((ext_vector_type(2))) float v2f;
typedef __attribute__((ext_vector_type(8))) float v8f;

#define NS      128      // samples
#define ND      24       // features
#define NF      276      // C(24,2) feature pairs
#define SPAIRS  8128     // C(128,2) sample pairs
#define NTILE   8        // 128/16 row tiles
#define NTP     36       // tile pairs with ti<=tj
#define MARGIN  0.5f

__global__ __launch_bounds__(32)
void rll_wmma_kernel(const int* __restrict__ labels,
                     const float* __restrict__ feats,
                     float* __restrict__ outW,
                     int* __restrict__ outIdx) {
    const int lane = threadIdx.x;            // wave32: one wave per block

    // Decode (feature pair f -> (p,q)) and (tile pair t -> (ti,tj), ti<=tj)
    int f = blockIdx.x / NTP;
    int t = blockIdx.x % NTP;

    int p = 0, rem = f, cnt = ND - 1;        // triu_indices(24,1) ordering
    while (rem >= cnt) { rem -= cnt; ++p; --cnt; }
    const int q = p + 1 + rem;

    int ti = 0, r2 = t, c2 = NTILE;          // (0,0),(0,1)..(0,7),(1,1)..
    while (r2 >= c2) { r2 -= c2; ++ti; --c2; }
    const int tj = ti + r2;

    // Per-lane staging: lanes 0-15 carry the K=0,1 factors; lanes 16-31 are
    // zeroed so the unused K=2,3 slots of the 16x16x4 WMMA contribute nothing.
    const int  l16  = lane & 15;
    const int  i    = ti * 16 + l16;         // i-side sample for A
    const int  j    = tj * 16 + l16;         // j-side sample for B
    const float msk = (lane < 16) ? 1.0f : 0.0f;

    const float yi = (float)(2 * labels[i] - 1);
    const float yj = (float)(2 * labels[j] - 1);
    const float up = yi * feats[i * ND + p];   // lf[i,p]
    const float uq = yi * feats[i * ND + q];   // lf[i,q]
    const float vp = yj * feats[j * ND + p];   // lf[j,p]
    const float vq = yj * feats[j * ND + q];   // lf[j,q]
    const float bi = yi - MARGIN;              // rhs component for row i
    const float bj = yj - MARGIN;              // rhs component for row j

    // det(m,n)  = up_i*vq_j - uq_i*vp_j
    // num0(m,n) = bi_i*vq_j - uq_i*bj_j      (W0 numerator)
    // num1(m,n) = up_i*bj_j - bi_i*vp_j      (W1 numerator)
    v2f aDet = { up * msk, -uq * msk };  v2f bDet = { vq * msk, vp * msk };
    v2f aN0  = { bi * msk, -uq * msk };  v2f bN0  = { vq * msk, bj * msk };
    v2f aN1  = { up * msk, -bi * msk };  v2f bN1  = { bj * msk, vp * msk };

    v8f zero = {};
    v8f det = __builtin_amdgcn_wmma_f32_16x16x4_f32(false, aDet, false, bDet,
                                                    (short)0, zero, false, false);
    v8f n0  = __builtin_amdgcn_wmma_f32_16x16x4_f32(false, aN0,  false, bN0,
                                                    (short)0, zero, false, false);
    v8f n1  = __builtin_amdgcn_wmma_f32_16x16x4_f32(false, aN1,  false, bN1,
                                                    (short)0, zero, false, false);

    // C/D layout: VGPR r -> M = r + 8*(lane>=16), N = lane&15.
    const int mBase = (lane < 16) ? 0 : 8;
    #pragma unroll
    for (int r = 0; r < 8; ++r) {
        const int mi = ti * 16 + mBase + r;   // global i
        const int nj = tj * 16 + l16;         // global j
        if (mi < nj) {                        // upper-triangle sample pairs only
            // s = triu pair index for (mi, nj), N=128
            const int s   = mi * (2 * NS - mi - 1) / 2 + (nj - mi - 1);
            const long long row = (long long)f * SPAIRS + s;
            const float d  = det[r];
            // hardware v_rcp_f32 (~1 ulp) instead of IEEE div expansion;
            // guard folds to v_cndmask (degenerate det -> w = 0)
            const float iv = (fabsf(d) > 1e-30f) ? __builtin_amdgcn_rcpf(d) : 0.0f;
            outW[row * 2 + 0]   = n0[r] * iv;
            outW[row * 2 + 1]   = n1[r] * iv;
            outIdx[row * 2 + 0] = p;
            outIdx[row * 2 + 1] = q;
        }
    }
}

extern "C" void kernel_launch(void* const* d_in, const int* in_sizes, int n_in,
                              void* d_out, int out_size, void* d_ws, size_t ws_size,
                              hipStream_t stream) {
    const int*   labels = (const int*)d_in[0];     // (128,) int32
    const float* feats  = (const float*)d_in[1];   // (128,24) f32
    float* outW   = (float*)d_out;                               // W: F*S*2 f32
    int*   outIdx = (int*)((float*)d_out + (size_t)NF * SPAIRS * 2); // col_indices: F*S*2 i32

    dim3 grid(NF * NTP);   // 276 * 36 = 9936 waves, one 16x16 (i,j) tile each
    dim3 block(32);        // one wave32 per block; EXEC all-1s at WMMA
    rll_wmma_kernel<<<grid, block, 0, stream>>>(labels, feats, outW, outIdx);
}